// AENN_73005854097729
// MI455X (gfx1250) — compile-verified
//
#include <hip/hip_runtime.h>

typedef float v2f __attribute__((ext_vector_type(2)));
typedef float v8f __attribute__((ext_vector_type(8)));
typedef unsigned int v4u __attribute__((ext_vector_type(4)));
typedef int v4i __attribute__((ext_vector_type(4)));
typedef int v8i __attribute__((ext_vector_type(8)));

#define DIM 1024
#define XS_STRIDE 1028   // pad: 16B-aligned rows, conflict-free WMMA frag loads

// ---------------------------------------------------------------------------
// Kernel A: build W = U^T (W[k][i] = U[i][k]) by simulating the circuit on
// each basis vector. One block per basis index, state kept in LDS (fp32 re/im).
// ---------------------------------------------------------------------------
__device__ __forceinline__ void apply_rot(float* re, float* im,
                                          float theta, float phi, float alpha,
                                          int j, int tid) {
  const float ct = cosf(theta), st = sinf(theta);
  const float sa = sinf(alpha);
  const float nx = sa * cosf(phi), ny = sa * sinf(phi), nz = cosf(alpha);
  const float snx = st * nx, sny = st * ny, snz = st * nz;
  const int b = 9 - j;
  __syncthreads();
  for (int p = tid; p < 512; p += 256) {
    const int i0 = ((p >> b) << (b + 1)) | (p & ((1 << b) - 1));
    const int i1 = i0 | (1 << b);
    const float r0 = re[i0], q0 = im[i0];
    const float r1 = re[i1], q1 = im[i1];
    // U00=(c,-s nz) U01=(-s ny,-s nx) U10=(s ny,-s nx) U11=(c, s nz)
    re[i0] = ct * r0 + snz * q0 - sny * r1 + snx * q1;
    im[i0] = ct * q0 - snz * r0 - sny * q1 - snx * r1;
    re[i1] = sny * r0 + snx * q0 + ct * r1 - snz * q1;
    im[i1] = sny * q0 - snx * r0 + ct * q1 + snz * r1;
  }
}

__device__ __forceinline__ void apply_rxx(float* re, float* im,
                                          float th, int j, int tid) {
  const float c = cosf(th), s = sinf(th);
  const int b = 9 - j;
  const int mask = (1 << b) | (1 << (b - 1));
  __syncthreads();
  for (int p = tid; p < 512; p += 256) {
    const int i0 = ((p >> b) << (b + 1)) | (p & ((1 << b) - 1));
    const int i1 = i0 ^ mask;
    const float r0 = re[i0], q0 = im[i0];
    const float r1 = re[i1], q1 = im[i1];
    // psi' = cos(th)*psi - i sin(th)*flip(psi)
    re[i0] = c * r0 + s * q1;
    im[i0] = c * q0 - s * r1;
    re[i1] = c * r1 + s * q0;
    im[i1] = c * q1 - s * r0;
  }
}

__global__ __launch_bounds__(256) void build_unitary(const float* __restrict__ qw,
                                                     float* __restrict__ Wre,
                                                     float* __restrict__ Wim) {
  __shared__ float re[DIM];
  __shared__ float im[DIM];
  const int tid = threadIdx.x;
  const int col = blockIdx.x;  // basis index k
  for (int i = tid; i < DIM; i += 256) {
    re[i] = (i == col) ? 1.0f : 0.0f;
    im[i] = 0.0f;
  }
  for (int l = 0; l < 4; ++l) {
    for (int j = 0; j < 10; ++j) {
      const float* r = qw + (l * 10 + j) * 3;
      apply_rot(re, im, r[0], r[1], r[2], j, tid);
    }
    for (int j = 0; j < 9; ++j)
      apply_rxx(re, im, qw[150 + l * 9 + j], j, tid);
  }
  for (int j = 0; j < 10; ++j) {
    const float* r = qw + (40 + j) * 3;
    apply_rot(re, im, r[0], r[1], r[2], j, tid);
  }
  __syncthreads();
  for (int i = tid; i < DIM; i += 256) {
    Wre[col * DIM + i] = re[i];
    Wim[col * DIM + i] = im[i];
  }
}

// ---------------------------------------------------------------------------
// Kernel B: out[r,o] = sum_i (Re[r,i]^2 + Im[r,i]^2) * kernel[i,o] + bias[o]
// where Re/Im = x @ W via V_WMMA_F32_16X16X4_F32. One block = 16 rows of x.
// 8 waves, each wave owns 128 amplitude columns (8 N-tiles of 16).
// kernel (1024x10) staged into LDS via the Tensor Data Mover.
// ---------------------------------------------------------------------------
__global__ __launch_bounds__(256) void qgemm_wmma(const float* __restrict__ x,
                                                  const float* __restrict__ Wre,
                                                  const float* __restrict__ Wim,
                                                  const float* __restrict__ kernel,
                                                  const float* __restrict__ bias,
                                                  float* __restrict__ out) {
  extern __shared__ float smem[];
  float* xs    = smem;                       // 16 x XS_STRIDE
  float* probs = xs + 16 * XS_STRIDE;        // 16 x XS_STRIDE
  float* kern  = probs + 16 * XS_STRIDE;     // 1024 x 10

  const int tid = threadIdx.x;
  const int r0 = blockIdx.x * 16;

#if __has_builtin(__builtin_amdgcn_tensor_load_to_lds)
  // TDM: DMA kernel matrix (10240 f32, one contiguous line) into LDS while the
  // vector pipes stage the x tile. Issued once, by wave 0 only.
  if (tid < 32) {
    const unsigned kern_lds = (unsigned)(size_t)kern;            // LDS offset = flat addr[31:0]
    const unsigned long long ga = (unsigned long long)(size_t)kernel;
    v4u g0;
    g0[0] = 1u;                                    // count=1 (valid user descriptor)
    g0[1] = kern_lds;                              // lds_addr (bytes)
    g0[2] = (unsigned)(ga & 0xffffffffu);          // global_addr[31:0]
    g0[3] = (unsigned)((ga >> 32) & 0x01ffffffu)   // global_addr[56:32]
            | 0x80000000u;                         // type=2 ("image") in bits 127:126
    v8i g1;
    g1[0] = 0x20000;                               // data_size=2 (4B); no mask/pad/iterate
    g1[1] = (int)(10240u << 16);                   // tensor_dim0[15:0] in bits 63:48
    g1[2] = (int)(1u << 16);                       // tensor_dim0[31:16]=0 | tensor_dim1=1
    g1[3] = (int)(10240u << 16);                   // tile_dim0=10240 in bits 127:112
    g1[4] = 0;                                     // tile_dim1=0, tile_dim2=0 (unused)
    g1[5] = 10240;                                 // tensor_dim0_stride[31:0]
    g1[6] = 0;                                     // stride hi / dim1_stride lo
    g1[7] = 0;
    const v4i gz4 = {0, 0, 0, 0};
    const v8i gz8 = {0, 0, 0, 0, 0, 0, 0, 0};
    __builtin_amdgcn_tensor_load_to_lds(g0, g1, gz4, gz4, gz8, 0);
  }
#else
  for (int idx = tid; idx < 1024 * 10; idx += 256) kern[idx] = kernel[idx];
#endif

  // Stage 16x1024 x-tile into LDS (float4, rows padded to XS_STRIDE).
  const float4* x4 = (const float4*)(x + (size_t)r0 * DIM);
  for (int idx = tid; idx < 16 * 256; idx += 256) {
    const int row = idx >> 8, c4 = idx & 255;
    float4 v = x4[row * 256 + c4];
    *(float4*)&xs[row * XS_STRIDE + c4 * 4] = v;
  }

#if __has_builtin(__builtin_amdgcn_tensor_load_to_lds)
  if (tid < 32) __builtin_amdgcn_s_wait_tensorcnt(0);
#endif
  __syncthreads();

  const int wave = tid >> 5;
  const int lane = tid & 31;
  const int half = lane >> 4;   // 0 or 1
  const int l16  = lane & 15;
  const int kb   = half << 1;   // K sub-offset within fragment (0 or 2)

  for (int t = 0; t < 8; ++t) {
    const int n0 = wave * 128 + t * 16;
    if (t < 7) {  // warm L1 with the next tile's B stream
      __builtin_prefetch(&Wre[n0 + 16 + l16], 0, 1);
      __builtin_prefetch(&Wim[n0 + 16 + l16], 0, 1);
    }
    v8f accR = {0.f, 0.f, 0.f, 0.f, 0.f, 0.f, 0.f, 0.f};
    v8f accI = {0.f, 0.f, 0.f, 0.f, 0.f, 0.f, 0.f, 0.f};
    const float* xrow = &xs[l16 * XS_STRIDE + kb];
    for (int k = 0; k < DIM; k += 4) {
      // A frag (16x4 of x): lane<16 -> K=0,1 ; lane>=16 -> K=2,3 (b64 LDS load)
      const v2f a = *(const v2f*)(xrow + k);
      const int krow = k + kb;
      v2f br, bi;
      br.x = Wre[krow * DIM + n0 + l16];
      br.y = Wre[(krow + 1) * DIM + n0 + l16];
      bi.x = Wim[krow * DIM + n0 + l16];
      bi.y = Wim[(krow + 1) * DIM + n0 + l16];
      accR = __builtin_amdgcn_wmma_f32_16x16x4_f32(false, a, false, br,
                                                   (short)0, accR, false, false);
      accI = __builtin_amdgcn_wmma_f32_16x16x4_f32(false, a, false, bi,
                                                   (short)0, accI, false, false);
    }
    // probs = Re^2 + Im^2, elementwise in C layout; scatter into LDS tile.
#pragma unroll
    for (int v = 0; v < 8; ++v) {
      const int m = v + half * 8;
      const float pr = accR[v] * accR[v] + accI[v] * accI[v];
      probs[m * XS_STRIDE + n0 + l16] = pr;
    }
  }
  __syncthreads();

  // Fixed-order contraction: 160 threads, one (row, output) each.
  if (tid < 160) {
    const int m = tid / 10, o = tid % 10;
    float acc = 0.0f;
    for (int n = 0; n < DIM; ++n)
      acc += probs[m * XS_STRIDE + n] * kern[n * 10 + o];
    out[(r0 + m) * 10 + o] = acc + bias[o];
  }
}

// ---------------------------------------------------------------------------
extern "C" void kernel_launch(void* const* d_in, const int* in_sizes, int n_in,
                              void* d_out, int out_size, void* d_ws, size_t ws_size,
                              hipStream_t stream) {
  const float* x    = (const float*)d_in[0];
  const float* qw   = (const float*)d_in[1];
  const float* kern = (const float*)d_in[2];
  const float* bias = (const float*)d_in[3];
  float* out = (float*)d_out;

  float* Wre = (float*)d_ws;               // 1024*1024 f32 = 4 MB
  float* Wim = Wre + (size_t)DIM * DIM;    // +4 MB

  build_unitary<<<DIM, 256, 0, stream>>>(qw, Wre, Wim);

  const size_t lds = (size_t)(2 * 16 * XS_STRIDE + 1024 * 10) * sizeof(float);
  qgemm_wmma<<<4096 / 16, 256, lds, stream>>>(x, Wre, Wim, kern, bias, out);
}